// GraphSAGE_24953759990543
// MI455X (gfx1250) — compile-verified
//
#include <hip/hip_runtime.h>

#define B_ 2
#define N_ 20000
#define F_ 256
#define E_ 640000
#define LDSTRIDE 260   // 260%4==0 (float4-aligned staging), 260%64==4 (conflict-free b64 frag reads)

typedef __attribute__((ext_vector_type(2))) float v2f;
typedef __attribute__((ext_vector_type(8))) float v8f;

// ---------------------------------------------------------------- zero ws ---
__global__ void sage_zero_kernel(float* __restrict__ ws, long n) {
    long i = (long)blockIdx.x * blockDim.x + threadIdx.x;
    long stride = (long)gridDim.x * blockDim.x;
    for (; i < n; i += stride) ws[i] = 0.0f;
}

// ---------------------------------------------------------------- degrees ---
__global__ void sage_degree_kernel(const long long* __restrict__ tgt,
                                   float* __restrict__ deg) {
    int i = blockIdx.x * blockDim.x + threadIdx.x;
    if (i < E_) {
        unsafeAtomicAdd(&deg[(int)tgt[i]], 1.0f);
    }
}

// ------------------------------------------------------- edge scatter-add ---
// One wave32 per edge: lanes sweep the 256 features coalesced, both batches.
__global__ __launch_bounds__(256) void sage_scatter_kernel(
        const float* __restrict__ x,
        const long long* __restrict__ src,
        const long long* __restrict__ tgt,
        float* __restrict__ agg) {
    int wave = (blockIdx.x * blockDim.x + threadIdx.x) >> 5;
    int lane = threadIdx.x & 31;
    if (wave >= E_) return;
    int s = (int)src[wave];
    int t = (int)tgt[wave];
    const float* xs0 = x + (size_t)s * F_;
    float* ag0 = agg + (size_t)t * F_;
#pragma unroll
    for (int b = 0; b < B_; ++b) {
        const float* xs = xs0 + (size_t)b * N_ * F_;
        float* ag = ag0 + (size_t)b * N_ * F_;
#pragma unroll
        for (int j = 0; j < F_ / 32; ++j) {
            int f = lane + j * 32;
            unsafeAtomicAdd(&ag[f], xs[f]);
        }
    }
}

// --------------------------------------------- fused dual-GEMM + bias+ReLU ---
// Block = 512 threads = 16 waves; block owns a 16-row tile (b, n0..n0+15);
// wave w owns output cols [16w, 16w+16). f32 WMMA 16x16x4, K=256, two GEMMs
// chained into one accumulator: acc = agg_norm*Wl^T + x*Wr^T.
__global__ __launch_bounds__(512) void sage_gemm_kernel(
        const float* __restrict__ x,
        const float* __restrict__ agg,
        const float* __restrict__ deg,
        const float* __restrict__ Wl,
        const float* __restrict__ bl,
        const float* __restrict__ Wr,
        float* __restrict__ out) {
    __shared__ float sAgg[16 * LDSTRIDE];
    __shared__ float sX[16 * LDSTRIDE];
    __shared__ float sInv[16];

    const int tid = threadIdx.x;
    const int b  = blockIdx.x / (N_ / 16);
    const int n0 = (blockIdx.x % (N_ / 16)) * 16;

    if (tid < 16) {
        sInv[tid] = 1.0f / fmaxf(deg[n0 + tid], 1.0f);
    }
    __syncthreads();

    // Stage 16x256 agg (normalized) and x tiles into LDS: 1024 float4, 2/thread.
    const float* aggRow = agg + ((size_t)b * N_ + n0) * F_;
    const float* xRow   = x   + ((size_t)b * N_ + n0) * F_;
    for (int i = tid; i < 1024; i += 512) {
        int r = i >> 6;           // row 0..15
        int f = (i & 63) << 2;    // col 0..252 step 4
        float4 va = *(const float4*)&aggRow[r * F_ + f];
        float inv = sInv[r];
        va.x *= inv; va.y *= inv; va.z *= inv; va.w *= inv;
        *(float4*)&sAgg[r * LDSTRIDE + f] = va;
        *(float4*)&sX[r * LDSTRIDE + f] = *(const float4*)&xRow[r * F_ + f];
    }
    __syncthreads();

    const int lane = tid & 31;
    const int wv   = tid >> 5;                 // col tile 0..15
    const int m    = lane & 15;                // A-frag row
    const int kk   = (lane >> 4) << 1;         // A/B frag K pair: 0 or 2
    const int ncol = wv * 16 + (lane & 15);    // B-frag / output column

    const float* wl = Wl + (size_t)ncol * F_;  // W is [F_OUT][F_IN] row-major
    const float* wr = Wr + (size_t)ncol * F_;
    __builtin_prefetch(wl, 0, 0);              // global_prefetch_b8
    __builtin_prefetch(wr, 0, 0);
    const float* pa = &sAgg[m * LDSTRIDE + kk];
    const float* px = &sX[m * LDSTRIDE + kk];

    v8f acc = {};
#pragma unroll 4
    for (int k = 0; k < F_; k += 4) {
        v2f aA = *(const v2f*)(pa + k);            // ds_load_b64
        v2f bA = *(const v2f*)(wl + k + kk);       // global_load_b64
        acc = __builtin_amdgcn_wmma_f32_16x16x4_f32(
                false, aA, false, bA, (short)0, acc, false, false);
        v2f aX = *(const v2f*)(px + k);
        v2f bX = *(const v2f*)(wr + k + kk);
        acc = __builtin_amdgcn_wmma_f32_16x16x4_f32(
                false, aX, false, bX, (short)0, acc, false, false);
    }

    // C/D layout: VGPR r -> row r (lanes 0-15) / row r+8 (lanes 16-31); col = lane&15.
    const float bias = bl[ncol];
    const int rbase = (lane >> 4) << 3;
    float* op = out + ((size_t)b * N_ + n0) * F_ + ncol;
#pragma unroll
    for (int r = 0; r < 8; ++r) {
        float v = acc[r] + bias;
        op[(size_t)(rbase + r) * F_] = fmaxf(v, 0.0f);
    }
}

// ------------------------------------------------------------------ launch ---
extern "C" void kernel_launch(void* const* d_in, const int* in_sizes, int n_in,
                              void* d_out, int out_size, void* d_ws, size_t ws_size,
                              hipStream_t stream) {
    const float*     x    = (const float*)d_in[0];       // [B,N,F]
    const long long* eidx = (const long long*)d_in[1];   // [2,E] int64
    const float*     Wl   = (const float*)d_in[2];       // [F,F]
    const float*     bl   = (const float*)d_in[3];       // [F]
    const float*     Wr   = (const float*)d_in[4];       // [F,F]
    float*           out  = (float*)d_out;               // [B,N,F]

    // Workspace: agg [B,N,F] (~41 MB) followed by deg [N].
    float* agg = (float*)d_ws;
    float* deg = agg + (size_t)B_ * N_ * F_;

    const long zn = (long)B_ * N_ * F_ + N_;
    sage_zero_kernel<<<2048, 256, 0, stream>>>((float*)d_ws, zn);
    sage_degree_kernel<<<(E_ + 255) / 256, 256, 0, stream>>>(eidx + E_, deg);
    sage_scatter_kernel<<<E_ / 8, 256, 0, stream>>>(x, eidx, eidx + E_, agg);
    sage_gemm_kernel<<<(B_ * N_) / 16, 512, 0, stream>>>(x, agg, deg, Wl, bl, Wr, out);
}